// QuantumLayer_18760417149183
// MI455X (gfx1250) — compile-verified
//
#include <hip/hip_runtime.h>
#include <hip/hip_bf16.h>

typedef __attribute__((ext_vector_type(2))) float v2f;
typedef __attribute__((ext_vector_type(8))) float v8f;

#define NQ 4
#define NL 2

// ---------------------------------------------------------------------------
// Setup kernel (1 block): build W = product of StronglyEntanglingLayers gates
// (16x16 complex), then M_q = Re(W^dagger Z_q W)  (4 real symmetric 16x16).
// Qubit q <-> bit (3-q) of the flattened basis index (C-order of (2,2,2,2)).
// ---------------------------------------------------------------------------
__global__ void build_quadforms(const float* __restrict__ w, float* __restrict__ M) {
    __shared__ float Wre[16][16];   // [basis k][column t]
    __shared__ float Wim[16][16];
    const int tid = threadIdx.x;

    if (tid < 16) {
        float re[16], im[16];
#pragma unroll
        for (int k = 0; k < 16; ++k) { re[k] = (k == tid) ? 1.f : 0.f; im[k] = 0.f; }

#pragma unroll
        for (int l = 0; l < NL; ++l) {
            // Rot(phi, theta, omega) on each wire
#pragma unroll
            for (int q = 0; q < NQ; ++q) {
                const float phi = w[(l * NQ + q) * 3 + 0];
                const float th  = w[(l * NQ + q) * 3 + 1];
                const float om  = w[(l * NQ + q) * 3 + 2];
                const float c = cosf(0.5f * th), s = sinf(0.5f * th);
                const float apo = 0.5f * (phi + om), amo = 0.5f * (phi - om);
                // U00 = e^{-i*apo} c ; U01 = -e^{+i*amo} s
                // U10 = e^{-i*amo} s ; U11 =  e^{+i*apo} c
                const float u00r =  cosf(apo) * c, u00i = -sinf(apo) * c;
                const float u01r = -cosf(amo) * s, u01i = -sinf(amo) * s;
                const float u10r =  cosf(amo) * s, u10i = -sinf(amo) * s;
                const float u11r =  cosf(apo) * c, u11i =  sinf(apo) * c;
                const int mask = 1 << (3 - q);
#pragma unroll
                for (int k = 0; k < 16; ++k) {
                    if (k & mask) continue;
                    const int k1 = k | mask;
                    const float a0r = re[k],  a0i = im[k];
                    const float a1r = re[k1], a1i = im[k1];
                    re[k]  = u00r * a0r - u00i * a0i + u01r * a1r - u01i * a1i;
                    im[k]  = u00r * a0i + u00i * a0r + u01r * a1i + u01i * a1r;
                    re[k1] = u10r * a0r - u10i * a0i + u11r * a1r - u11i * a1i;
                    im[k1] = u10r * a0i + u10i * a0r + u11r * a1i + u11i * a1r;
                }
            }
            // CNOT ring with range r = (l % (n-1)) + 1
            const int rr = (l % (NQ - 1)) + 1;
#pragma unroll
            for (int q = 0; q < NQ; ++q) {
                const int cm = 1 << (3 - q);
                const int tm = 1 << (3 - ((q + rr) % NQ));
#pragma unroll
                for (int k = 0; k < 16; ++k) {
                    if ((k & cm) && !(k & tm)) {
                        const int k1 = k | tm;
                        float t;
                        t = re[k]; re[k] = re[k1]; re[k1] = t;
                        t = im[k]; im[k] = im[k1]; im[k1] = t;
                    }
                }
            }
        }
#pragma unroll
        for (int k = 0; k < 16; ++k) { Wre[k][tid] = re[k]; Wim[k][tid] = im[k]; }
    }
    __syncthreads();

    // M_q[i][j] = sum_k z_q(k) * (Wre[k][i]Wre[k][j] + Wim[k][i]Wim[k][j])
    const int i = tid >> 4, j = tid & 15;
#pragma unroll
    for (int q = 0; q < NQ; ++q) {
        float acc = 0.f;
#pragma unroll
        for (int k = 0; k < 16; ++k) {
            const float z = ((k >> (3 - q)) & 1) ? -1.f : 1.f;
            acc += z * (Wre[k][i] * Wre[k][j] + Wim[k][i] * Wim[k][j]);
        }
        M[q * 256 + i * 16 + j] = acc;
    }
}

// ---------------------------------------------------------------------------
// Batch kernel. Per 16-sample tile:
//   build V (16 samples x 16 basis, real) -> LDS (Vt needed for B operand),
//   Pt_q = M_q x Vt  via V_WMMA_F32_16X16X4_F32 (K=16 as 4 chunks),
//   y[s][q] = sum_i V[s][i] * Pt_q[i][s]
// In the D/C layout (col s = lane&15, rows i = r + 8h) the 8 accumulator
// VGPRs pair elementwise with the 8 V values this lane computed locally,
// so the reduction is 8 FMAs + ONE cross-half xor-16 exchange per q.
// ---------------------------------------------------------------------------
__global__ __launch_bounds__(256) void qf_eval(const float* __restrict__ x,
                                               const float* __restrict__ M,
                                               float* __restrict__ out,
                                               int nsamp) {
    __shared__ float vlds[8][16 * 17];          // per-wave 16x16 V tile, padded
    const int lane = threadIdx.x & 31;
    const int wid  = threadIdx.x >> 5;
    const int m    = lane & 15;                 // row i (A) / sample s (B,C layouts)
    const int h    = lane >> 4;                 // wave half
    float* lds = &vlds[wid][0];

    // A-layout operands: M_q rows. fp32 A 16x4: M = lane&15,
    // lane<16 -> (K0,K1), lane>=16 -> (K2,K3). Rows are contiguous -> b64 loads.
    v2f amat[4][4];
#pragma unroll
    for (int q = 0; q < 4; ++q)
#pragma unroll
        for (int k = 0; k < 4; ++k)
            amat[q][k] = *(const v2f*)(M + q * 256 + m * 16 + 4 * k + 2 * h);

    const int ntiles = nsamp >> 4;
    const int gw = blockIdx.x * 8 + wid;
    const int nw = gridDim.x * 8;

    for (int tile = gw; tile < ntiles; tile += nw) {
        const int s = tile * 16 + m;
        const float4 xa = ((const float4*)x)[s];
        const float c0 = __cosf(0.5f * xa.x), s0 = __sinf(0.5f * xa.x);
        const float c1 = __cosf(0.5f * xa.y), s1 = __sinf(0.5f * xa.y);
        const float c2 = __cosf(0.5f * xa.z), s2 = __sinf(0.5f * xa.z);
        const float c3 = __cosf(0.5f * xa.w), s3 = __sinf(0.5f * xa.w);

        // v[i] = prod_q (bit_{3-q}(i) ? sin : cos); this lane owns sample m,
        // basis 8h..8h+7 — kept in registers AND staged to LDS for the B read.
        const float vq0 = h ? s0 : c0;          // bit3 == h over this lane's range
        float vreg[8];
#pragma unroll
        for (int ii = 0; ii < 8; ++ii) {
            float v = vq0;
            v *= ((ii >> 2) & 1) ? s1 : c1;
            v *= ((ii >> 1) & 1) ? s2 : c2;
            v *= (ii & 1)        ? s3 : c3;
            vreg[ii] = v;
            lds[m * 17 + 8 * h + ii] = v;
        }
        __asm volatile("s_wait_dscnt 0" ::: "memory");   // same-wave LDS RAW

        // B-layout Vt chunks: B[j][s], rows j = 4k+2h+{0,1}, col s = lane&15
        v2f bv[4];
#pragma unroll
        for (int k = 0; k < 4; ++k) {
            bv[k][0] = lds[m * 17 + 4 * k + 2 * h];
            bv[k][1] = lds[m * 17 + 4 * k + 2 * h + 1];
        }

        float y[4];
#pragma unroll
        for (int q = 0; q < 4; ++q) {
            v8f acc = {};
#pragma unroll
            for (int k = 0; k < 4; ++k)
                acc = __builtin_amdgcn_wmma_f32_16x16x4_f32(
                    false, amat[q][k], false, bv[k], (short)0, acc, false, false);

            // local dot: rows 8h..8h+7 of column m pair with vreg
            float t = 0.f;
#pragma unroll
            for (int r = 0; r < 8; ++r) t = fmaf(acc[r], vreg[r], t);
            // add the other half's partial (rows of the other 8-block):
            // ds_swizzle xor-16 within the wave (single instruction)
            const float to = __int_as_float(
                __builtin_amdgcn_ds_swizzle(__float_as_int(t), 0x401F));
            y[q] = t + to;
        }

        if (h == 0)   // one coalesced 16B store per sample: out[s][0..3]
            ((float4*)out)[tile * 16 + m] = make_float4(y[0], y[1], y[2], y[3]);
    }
}

extern "C" void kernel_launch(void* const* d_in, const int* in_sizes, int n_in,
                              void* d_out, int out_size, void* d_ws, size_t ws_size,
                              hipStream_t stream) {
    const float* x = (const float*)d_in[0];     // (B, 4) f32
    const float* w = (const float*)d_in[1];     // (2, 4, 3) f32
    float* out = (float*)d_out;                 // (B, 4) f32
    float* M   = (float*)d_ws;                  // 4 * 16 * 16 f32 quadratic forms
    const int nsamp = in_sizes[0] / NQ;

    build_quadforms<<<1, 256, 0, stream>>>(w, M);

    const int ntiles = nsamp >> 4;
    int blocks = (ntiles + 63) >> 6;            // 8 waves/block, ~8 tiles/wave
    if (blocks < 1) blocks = 1;
    if (blocks > 4096) blocks = 4096;
    qf_eval<<<blocks, 256, 0, stream>>>(x, M, out, nsamp);
}